// MultiresolutionRenderer_59184649339617
// MI455X (gfx1250) — compile-verified
//
#include <hip/hip_runtime.h>
#include <math.h>

// ---------------- problem constants (from the reference) ----------------
#define CH     5
#define MHW    512
#define NIM    2048      // good_fft_size(4*512)
#define NPM    192       // good_fft_size(4*41)
#define NPO    256       // good_fft_size(4*61)
#define TT     257       // T = min(n_im, n_po) + 1
#define NKX    544       // model-spectrum columns computed (covers <= ~536)
#define RMIN   480       // model-spectrum row window start (covers 493..1556)
#define NR     1088      // model-spectrum row window size
#define PM_P   41
#define PO_P   61
#define PM_NKX 97
#define PO_NKX 129
#define ZN     144       // padded columns for inverse transform (129 -> 144)
#define EM     224       // padded output extent (220 -> 224)
#define OBS    220
#define PI_D   3.14159265358979323846

typedef float v2f __attribute__((ext_vector_type(2)));
typedef float v8f __attribute__((ext_vector_type(8)));

// f32 16x16x4 WMMA (codegen-confirmed round 1)
__device__ __forceinline__ v8f wmma4(v2f a, v2f b, v8f c) {
  return __builtin_amdgcn_wmma_f32_16x16x4_f32(false, a, false, b, (short)0, c,
                                               false, false);
}

// ------------------------------------------------------------------------
// DFT GEMM: 4 waves/block, block tile 64(M) x 64(N), K chunked by 16 with
// the B chunk staged in LDS (pair-interleaved: float2 {B[2p][n],B[2p+1][n]}).
// Each wave computes a 16x64 strip (4 accumulators), A fragments are
// contiguous b64 loads from planar row-major A.
// MODE 0: real A x complex B -> complex D
// MODE 1: complex A x complex B -> complex D
// MODE 2: complex A x (cos,-sin) real pair -> real D
// PAIR: store D pair-interleaved (when D is the next GEMM's B operand)
// ------------------------------------------------------------------------
template <int MODE, bool PAIR>
__global__ __launch_bounds__(128) void dft_gemm(
    const float* __restrict__ Ar_, const float* __restrict__ Ai_,
    const float2* __restrict__ Brp_, const float2* __restrict__ Bip_,
    float* __restrict__ Dr_, float* __restrict__ Di_,
    int M, int N, int K,
    long sA, long sB, long sD,   // batch strides: A floats, B float2, D native units
    int Mstore, int Nstore, int ldD) {
  __shared__ float2 sBr[8][64];   // 8 K-pairs x 64 cols
  __shared__ float2 sBi[8][64];

  const int lane = threadIdx.x & 31;
  const int wave = threadIdx.x >> 5;
  const int bN0 = blockIdx.x << 6;
  const int m0  = (blockIdx.y << 6) + (wave << 4);
  const int b   = blockIdx.z;
  const int mrc = min(m0 + (lane & 15), M - 1);  // clamped A row (stores guarded)
  const int kb  = (lane >> 4) << 1;              // 0 or 2

  const float*  Ar  = Ar_ + (long)b * sA;
  const float*  Ai  = (MODE != 0) ? (Ai_ + (long)b * sA) : nullptr;
  const float2* Brp = Brp_ + (long)b * sB;
  const float2* Bip = Bip_ + (long)b * sB;

  v8f accR[4], accI[4];
#pragma unroll
  for (int t = 0; t < 4; ++t) {
    accR[t] = (v8f){0.f, 0.f, 0.f, 0.f, 0.f, 0.f, 0.f, 0.f};
    accI[t] = (v8f){0.f, 0.f, 0.f, 0.f, 0.f, 0.f, 0.f, 0.f};
  }

  for (int k0 = 0; k0 < K; k0 += 16) {
    // ---- stage B chunk (16 K values = 8 pairs, 64 cols) into LDS ----
    const int k2_0 = k0 >> 1;
#pragma unroll
    for (int e = threadIdx.x; e < 512; e += 128) {
      int row = e >> 6, col = e & 63;
      int gc = bN0 + col;
      float2 vr = make_float2(0.f, 0.f), vi = make_float2(0.f, 0.f);
      if (gc < N) {
        long o = (long)(k2_0 + row) * N + gc;
        vr = Brp[o];
        vi = Bip[o];
        if (k0 + 16 < K) __builtin_prefetch(&Brp[o + 8L * N], 0, 1);
      }
      sBr[row][col] = vr;
      sBi[row][col] = vi;
    }
    __syncthreads();

    // ---- 4 K-steps of 4, 4 N-subtiles each ----
#pragma unroll
    for (int ks = 0; ks < 4; ++ks) {
      const int kk = k0 + (ks << 2) + kb;          // global K for A fragment
      const int p  = (ks << 1) + (lane >> 4);      // LDS pair row
      v2f a_re = *(const v2f*)(Ar + (long)mrc * K + kk);
      v2f a_im = {0.f, 0.f}, na_im = {0.f, 0.f};
      if (MODE != 0) {
        a_im = *(const v2f*)(Ai + (long)mrc * K + kk);
        na_im = -a_im;
      }
#pragma unroll
      for (int nt = 0; nt < 4; ++nt) {
        v2f br = *(const v2f*)&sBr[p][(nt << 4) + (lane & 15)];
        v2f bi = *(const v2f*)&sBi[p][(nt << 4) + (lane & 15)];
        if (MODE == 0) {
          accR[nt] = wmma4(a_re, br, accR[nt]);
          accI[nt] = wmma4(a_re, bi, accI[nt]);
        } else if (MODE == 1) {
          accR[nt] = wmma4(a_re, br, accR[nt]);
          accR[nt] = wmma4(na_im, bi, accR[nt]);   // - Ai*Bi
          accI[nt] = wmma4(a_re, bi, accI[nt]);
          accI[nt] = wmma4(a_im, br, accI[nt]);
        } else {
          accR[nt] = wmma4(a_re, br, accR[nt]);    // Ar*cos
          accR[nt] = wmma4(a_im, bi, accR[nt]);    // Ai*(-sin)
        }
      }
    }
    __syncthreads();
  }

  // ---- store ----
  const int rbase = m0 + ((lane >> 4) << 3);       // even
#pragma unroll
  for (int nt = 0; nt < 4; ++nt) {
    const int nc = bN0 + (nt << 4) + (lane & 15);
    if (nc >= Nstore) continue;
    if (PAIR) {
      float2* DrP = (float2*)Dr_ + (long)b * sD;
      float2* DiP = (float2*)Di_ + (long)b * sD;
#pragma unroll
      for (int j = 0; j < 8; j += 2) {
        int r = rbase + j;
        if (r < Mstore) {
          long o = (long)(r >> 1) * ldD + nc;
          DrP[o] = make_float2(accR[nt][j], accR[nt][j + 1]);
          DiP[o] = make_float2(accI[nt][j], accI[nt][j + 1]);
        }
      }
    } else {
      float* Dr = Dr_ + (long)b * sD;
      float* Di = (MODE != 2) ? (Di_ + (long)b * sD) : nullptr;
#pragma unroll
      for (int j = 0; j < 8; ++j) {
        int r = rbase + j;
        if (r < Mstore) {
          long o = (long)r * ldD + nc;
          Dr[o] = accR[nt][j];
          if (MODE != 2) Di[o] = accI[nt][j];
        }
      }
    }
  }
}

// ---------------- twiddle initialization ----------------
// stage-1 B: W1[x][kx] = exp(-2pi i kx (x-256)/2048), pair-interleaved along x
__global__ void init_w1(float2* Wr, float2* Wi) {
  int idx = blockIdx.x * blockDim.x + threadIdx.x;
  if (idx >= (MHW / 2) * NKX) return;
  int x2 = idx / NKX, kx = idx % NKX;
  double s0, c0, s1, c1;
  sincos(-2.0 * PI_D * (double)kx * (double)(2 * x2 - 256) / (double)NIM, &s0, &c0);
  sincos(-2.0 * PI_D * (double)kx * (double)(2 * x2 + 1 - 256) / (double)NIM, &s1, &c1);
  Wr[idx] = make_float2((float)c0, (float)c1);
  Wi[idx] = make_float2((float)s0, (float)s1);
}
// stage-2 A (planar): T2[r][y] = exp(-2pi i (r+RMIN-1024)(y-256)/2048)
__global__ void init_t2(float* Tr, float* Ti) {
  int idx = blockIdx.x * blockDim.x + threadIdx.x;
  if (idx >= NR * MHW) return;
  int r = idx / MHW, y = idx % MHW;
  double ang = -2.0 * PI_D * (double)(r + RMIN - NIM / 2) * (double)(y - 256) / (double)NIM;
  double s, c; sincos(ang, &s, &c);
  Tr[idx] = (float)c; Ti[idx] = (float)s;
}
// inverse-row A (planar, fftshift+crop folded): E1M[yy][q] = exp(+2pi i q (yy+146)/256)/256
__global__ void init_e1m(float* Er, float* Ei) {
  int idx = blockIdx.x * blockDim.x + threadIdx.x;
  if (idx >= EM * NPO) return;
  int yy = idx / NPO, q = idx % NPO;
  double ang = 2.0 * PI_D * (double)q * (double)((yy + 146) % NPO) / (double)NPO;
  double s, c; sincos(ang, &s, &c);
  Er[idx] = (float)(c / NPO); Ei[idx] = (float)(s / NPO);
}
// final B pair: Cc[x][xx] = w cos(2pi x (xx+146)/256)/256 ; Cs = -w sin(...)
__global__ void init_c2(float2* Ccp, float2* Csp) {
  int idx = blockIdx.x * blockDim.x + threadIdx.x;
  if (idx >= (ZN / 2) * EM) return;
  int x2 = idx / EM, xx = idx % EM;
  float c01[2], s01[2];
#pragma unroll
  for (int h = 0; h < 2; ++h) {
    int x = 2 * x2 + h;
    float cv = 0.f, sv = 0.f;
    if (x <= 128) {
      double w = (x == 0 || x == 128) ? 1.0 : 2.0;
      double s, c;
      sincos(2.0 * PI_D * (double)x * (double)(xx + 146) / (double)NPO, &s, &c);
      cv = (float)(w * c / NPO);
      sv = (float)(-w * s / NPO);
    }
    c01[h] = cv; s01[h] = sv;
  }
  Ccp[idx] = make_float2(c01[0], c01[1]);
  Csp[idx] = make_float2(s01[0], s01[1]);
}

// ---------------- small PSF spectra via separable direct DFT ----------------
__global__ void psf_rowdft(const float* __restrict__ psf, float2* __restrict__ tmp,
                           int P, int n, int off, int NK2, int total) {
  int idx = blockIdx.x * blockDim.x + threadIdx.x;
  if (idx >= total) return;
  int i = idx % NK2, cu = idx / NK2;
  double sr = 0.0, si = 0.0;
  for (int v = 0; v < P; ++v) {
    double s, c;
    sincos(-2.0 * PI_D * (double)i * (double)(v - off) / (double)n, &s, &c);
    double val = (double)psf[(long)cu * P + v];
    sr += val * c; si += val * s;
  }
  tmp[idx] = make_float2((float)sr, (float)si);
}
__global__ void psf_coldft(const float2* __restrict__ tmp, float2* __restrict__ out,
                           int P, int n, int off, int NK2, int NRo, int total) {
  int idx = blockIdx.x * blockDim.x + threadIdx.x;
  if (idx >= total) return;
  int i = idx % NK2, t = idx / NK2, r = t % NRo, ch = t / NRo;
  int ky = r - n / 2;                 // fftshift along -2 folded in
  double ar = 0.0, ai = 0.0;
  for (int u = 0; u < P; ++u) {
    double s, c;
    sincos(-2.0 * PI_D * (double)ky * (double)(u - off) / (double)n, &s, &c);
    float2 tv = tmp[(long)(ch * P + u) * NK2 + i];
    ar += tv.x * c - tv.y * s;
    ai += tv.x * s + tv.y * c;
  }
  out[idx] = make_float2((float)ar, (float)ai);
}

// ---------------- bilinear resample (reference-exact bounds) ----------------
__device__ __forceinline__ float2 ksampleI(const float2* base, int ld, int ny, int nx,
                                           int r, int c) {
  if (r < 0 || r >= ny || c < 0 || c >= nx) return make_float2(0.f, 0.f);
  return base[(long)r * ld + c];
}
__device__ __forceinline__ float2 kbilinearI(const float2* base, int ld, int ny, int nx,
                                             float rows, float cols) {
  int r0 = (int)floorf(rows), c0 = (int)floorf(cols);
  float dr = rows - (float)r0, dc = cols - (float)c0;
  float2 g00 = ksampleI(base, ld, ny, nx, r0, c0);
  float2 g10 = ksampleI(base, ld, ny, nx, r0 + 1, c0);
  float2 g01 = ksampleI(base, ld, ny, nx, r0, c0 + 1);
  float2 g11 = ksampleI(base, ld, ny, nx, r0 + 1, c0 + 1);
  float w00 = (1.f - dr) * (1.f - dc), w10 = dr * (1.f - dc);
  float w01 = (1.f - dr) * dc, w11 = dr * dc;
  return make_float2(g00.x * w00 + g10.x * w10 + g01.x * w01 + g11.x * w11,
                     g00.y * w00 + g10.y * w10 + g01.y * w01 + g11.y * w11);
}
__device__ __forceinline__ float2 ksampleP(const float* br, const float* bi, int ld,
                                           int ny, int nx, int wr0, int wnr,
                                           int r, int c) {
  if (r < 0 || r >= ny || c < 0 || c >= nx) return make_float2(0.f, 0.f);
  int rr = r - wr0;
  if (rr < 0 || rr >= wnr || c >= ld) return make_float2(0.f, 0.f);
  long o = (long)rr * ld + c;
  return make_float2(br[o], bi[o]);
}
__device__ __forceinline__ float2 kbilinearP(const float* br, const float* bi, int ld,
                                             int ny, int nx, int wr0, int wnr,
                                             float rows, float cols) {
  int r0 = (int)floorf(rows), c0 = (int)floorf(cols);
  float dr = rows - (float)r0, dc = cols - (float)c0;
  float2 g00 = ksampleP(br, bi, ld, ny, nx, wr0, wnr, r0, c0);
  float2 g10 = ksampleP(br, bi, ld, ny, nx, wr0, wnr, r0 + 1, c0);
  float2 g01 = ksampleP(br, bi, ld, ny, nx, wr0, wnr, r0, c0 + 1);
  float2 g11 = ksampleP(br, bi, ld, ny, nx, wr0, wnr, r0 + 1, c0 + 1);
  float w00 = (1.f - dr) * (1.f - dc), w10 = dr * (1.f - dc);
  float w01 = (1.f - dr) * dc, w11 = dr * dc;
  return make_float2(g00.x * w00 + g10.x * w10 + g01.x * w01 + g11.x * w11,
                     g00.y * w00 + g10.y * w10 + g01.y * w01 + g11.y * w11);
}

// fused resample of the three spectra + kfinal = kim/kpsf*kobs
__global__ void resample_kfinal(const float* __restrict__ F2r,
                                const float* __restrict__ F2i,
                                const float2* __restrict__ Pm,
                                const float2* __restrict__ Po,
                                float2* __restrict__ KF) {
  int idx = blockIdx.x * blockDim.x + threadIdx.x;
  if (idx >= CH * TT * PO_NKX) return;
  int i = idx % PO_NKX, j = (idx / PO_NKX) % TT, ch = idx / (PO_NKX * TT);
  const float KX = (float)i, KY = (float)(j - 129);   // ymin = -T//2 = -129
  float cs, sn; __sincosf(0.05f, &sn, &cs);

  // --- model image spectrum (window of 2048 x 1025 grid), phi + flip ---
  float sc = (float)NIM * 0.2f / ((float)TT * 0.4f);
  float fx = KX * sc, fy = KY * sc;
  float fx2 = cs * fx - sn * fy, fy2 = sn * fx + cs * fy;
  bool negI = fx2 < 0.f;
  float fxs = negI ? -fx2 : fx2, fys = negI ? -fy2 : fy2;
  float2 kim = kbilinearP(F2r + (long)ch * NR * NKX, F2i + (long)ch * NR * NKX,
                          NKX, NIM, NIM / 2 + 1, RMIN, NR,
                          fys + (float)(NIM / 2), fxs);
  if (negI) kim.y = -kim.y;

  // --- model PSF spectrum (192 x 97), phi + flip ---
  sc = (float)NPM * 0.2f / ((float)TT * 0.4f);
  fx = KX * sc; fy = KY * sc;
  fx2 = cs * fx - sn * fy; fy2 = sn * fx + cs * fy;
  bool negP = fx2 < 0.f;
  fxs = negP ? -fx2 : fx2; fys = negP ? -fy2 : fy2;
  float2 kpm = kbilinearI(Pm + (long)ch * NPM * PM_NKX, PM_NKX, NPM, PM_NKX,
                          fys + (float)(NPM / 2), fxs);
  if (negP) kpm.y = -kpm.y;

  // --- obs PSF spectrum (256 x 129), no rotation/flip ---
  sc = (float)NPO * 0.4f / ((float)TT * 0.4f);
  fx = KX * sc; fy = KY * sc;
  bool negO = fx < 0.f;
  fxs = negO ? -fx : fx; fys = negO ? -fy : fy;
  float2 kpo = kbilinearI(Po + (long)ch * NPO * PO_NKX, PO_NKX, NPO, PO_NKX,
                          fys + (float)(NPO / 2), fxs);
  if (negO) kpo.y = -kpo.y;

  // kfinal = (kim / kpm) * kpo
  float den = kpm.x * kpm.x + kpm.y * kpm.y;
  float2 t = make_float2(kim.x * kpm.x + kim.y * kpm.y,
                         kim.y * kpm.x - kim.x * kpm.y);
  KF[idx] = make_float2((t.x * kpo.x - t.y * kpo.y) / den,
                        (t.x * kpo.y + t.y * kpo.x) / den);
}

// hermitian wrap (row 256 folds onto row 0, Nyquist col zeroed) + ifftshift(-2),
// emitted pair-interleaved along q (B operand of the inverse GEMM), padded to 144 cols
__global__ void wrap_shift(const float2* __restrict__ KF,
                           float2* __restrict__ Zrp, float2* __restrict__ Zip) {
  int idx = blockIdx.x * blockDim.x + threadIdx.x;
  if (idx >= CH * (NPO / 2) * ZN) return;
  int x = idx % ZN, q2 = (idx / ZN) % (NPO / 2), ch = idx / (ZN * (NPO / 2));
  float2 v[2];
#pragma unroll
  for (int h = 0; h < 2; ++h) {
    float2 val = make_float2(0.f, 0.f);
    if (x < 128) {
      int r = (2 * q2 + h + 128) & 255;
      val = KF[((long)ch * TT + r) * PO_NKX + x];
      if (r == 0) {
        float2 w = KF[((long)ch * TT + 256) * PO_NKX + x];
        val.x += w.x; val.y += w.y;
      }
    }
    v[h] = val;
  }
  Zrp[idx] = make_float2(v[0].x, v[1].x);
  Zip[idx] = make_float2(v[0].y, v[1].y);
}

// ---------------- host orchestration ----------------
extern "C" void kernel_launch(void* const* d_in, const int* in_sizes, int n_in,
                              void* d_out, int out_size, void* d_ws, size_t ws_size,
                              hipStream_t stream) {
  const float* model = (const float*)d_in[0];   // [5][512][512]
  const float* psf_m = (const float*)d_in[1];   // [5][41][41]
  const float* psf_o = (const float*)d_in[2];   // [5][61][61]
  float* out = (float*)d_out;                   // [5][220][220]

  size_t off = 0;
  auto take = [&](size_t nbytes) {
    size_t cur = off;
    off += (nbytes + 255) & ~(size_t)255;
    return cur;
  };
  char* ws = (char*)d_ws;
  float2* W1rp = (float2*)(ws + take((size_t)(MHW / 2) * NKX * 8));
  float2* W1ip = (float2*)(ws + take((size_t)(MHW / 2) * NKX * 8));
  float*  T2r  = (float*)(ws + take((size_t)NR * MHW * 4));
  float*  T2i  = (float*)(ws + take((size_t)NR * MHW * 4));
  float2* F1rp = (float2*)(ws + take((size_t)CH * (MHW / 2) * NKX * 8));
  float2* F1ip = (float2*)(ws + take((size_t)CH * (MHW / 2) * NKX * 8));
  float*  F2r  = (float*)(ws + take((size_t)CH * NR * NKX * 4));
  float*  F2i  = (float*)(ws + take((size_t)CH * NR * NKX * 4));
  float2* tmpM = (float2*)(ws + take((size_t)CH * PM_P * PM_NKX * 8));
  float2* Pm   = (float2*)(ws + take((size_t)CH * NPM * PM_NKX * 8));
  float2* tmpO = (float2*)(ws + take((size_t)CH * PO_P * PO_NKX * 8));
  float2* Po   = (float2*)(ws + take((size_t)CH * NPO * PO_NKX * 8));
  float2* KF   = (float2*)(ws + take((size_t)CH * TT * PO_NKX * 8));
  float2* Zrp  = (float2*)(ws + take((size_t)CH * (NPO / 2) * ZN * 8));
  float2* Zip  = (float2*)(ws + take((size_t)CH * (NPO / 2) * ZN * 8));
  float*  E1Mr = (float*)(ws + take((size_t)EM * NPO * 4));
  float*  E1Mi = (float*)(ws + take((size_t)EM * NPO * 4));
  float*  B1r  = (float*)(ws + take((size_t)CH * EM * ZN * 4));
  float*  B1i  = (float*)(ws + take((size_t)CH * EM * ZN * 4));
  float2* C2cp = (float2*)(ws + take((size_t)(ZN / 2) * EM * 8));
  float2* C2sp = (float2*)(ws + take((size_t)(ZN / 2) * EM * 8));

  const int TB = 256;
  auto blocks = [&](long n) { return (unsigned)((n + TB - 1) / TB); };

  // twiddles
  init_w1<<<blocks((long)(MHW / 2) * NKX), TB, 0, stream>>>(W1rp, W1ip);
  init_t2<<<blocks((long)NR * MHW), TB, 0, stream>>>(T2r, T2i);
  init_e1m<<<blocks((long)EM * NPO), TB, 0, stream>>>(E1Mr, E1Mi);
  init_c2<<<blocks((long)(ZN / 2) * EM), TB, 0, stream>>>(C2cp, C2sp);

  // PSF spectra (separable direct DFT)
  psf_rowdft<<<blocks((long)CH * PM_P * PM_NKX), TB, 0, stream>>>(
      psf_m, tmpM, PM_P, NPM, 21, PM_NKX, CH * PM_P * PM_NKX);
  psf_coldft<<<blocks((long)CH * NPM * PM_NKX), TB, 0, stream>>>(
      tmpM, Pm, PM_P, NPM, 21, PM_NKX, NPM, CH * NPM * PM_NKX);
  psf_rowdft<<<blocks((long)CH * PO_P * PO_NKX), TB, 0, stream>>>(
      psf_o, tmpO, PO_P, NPO, 31, PO_NKX, CH * PO_P * PO_NKX);
  psf_coldft<<<blocks((long)CH * NPO * PO_NKX), TB, 0, stream>>>(
      tmpO, Po, PO_P, NPO, 31, PO_NKX, NPO, CH * NPO * PO_NKX);

  // stage 1: F1[y][kx] = sum_x model[y][x] * W1[x][kx]   (real x complex, pair-store)
  dft_gemm<0, true><<<dim3((NKX + 63) / 64, MHW / 64, CH), 128, 0, stream>>>(
      model, nullptr, W1rp, W1ip, (float*)F1rp, (float*)F1ip,
      MHW, NKX, MHW, (long)MHW * MHW, 0L, (long)(MHW / 2) * NKX,
      MHW, NKX, NKX);
  // stage 2: F2[r][kx] = sum_y T2[r][y] * F1[y][kx]      (complex x complex, planar)
  dft_gemm<1, false><<<dim3((NKX + 63) / 64, NR / 64, CH), 128, 0, stream>>>(
      T2r, T2i, F1rp, F1ip, F2r, F2i,
      NR, NKX, MHW, 0L, (long)(MHW / 2) * NKX, (long)NR * NKX,
      NR, NKX, NKX);

  // resample + kfinal, then hermitian wrap + ifftshift into pair-layout Z
  resample_kfinal<<<blocks((long)CH * TT * PO_NKX), TB, 0, stream>>>(F2r, F2i, Pm, Po, KF);
  wrap_shift<<<blocks((long)CH * (NPO / 2) * ZN), TB, 0, stream>>>(KF, Zrp, Zip);

  // inverse rows: B1[yy][x] = sum_q E1M[yy][q] * Z[q][x]  (complex x complex, planar)
  dft_gemm<1, false><<<dim3((ZN + 63) / 64, (EM + 63) / 64, CH), 128, 0, stream>>>(
      E1Mr, E1Mi, Zrp, Zip, B1r, B1i,
      EM, ZN, NPO, 0L, (long)(NPO / 2) * ZN, (long)EM * ZN,
      EM, ZN, ZN);
  // final: out[yy][xx] = Re(B1)*Cc + Im(B1)*Cs           (fused irfft + shift + crop)
  dft_gemm<2, false><<<dim3((EM + 63) / 64, (EM + 63) / 64, CH), 128, 0, stream>>>(
      B1r, B1i, C2cp, C2sp, out, nullptr,
      EM, EM, ZN, (long)EM * ZN, 0L, (long)OBS * OBS,
      OBS, OBS, OBS);
}